// LeeTanh_39822936769147
// MI455X (gfx1250) — compile-verified
//
#include <hip/hip_runtime.h>
#include <math.h>

// Lee-oscillator constants (match reference)
#define LEE_ITERS 49            // N - 1 scan steps; 49 = 7 * 7
#define LEE_K     50.0f
#define LOG2E_F   1.4426950408889634f

// ---------------------------------------------------------------------------
// gfx1250 hardware transcendentals.
// v_tanh_f32 is new on gfx1250 — one TRANS instruction instead of the ~20-op
// ocml polynomial. Guarded so we always emit the HW instruction on device.
// ---------------------------------------------------------------------------
__device__ __forceinline__ float dev_tanh(float x) {
#if defined(__AMDGCN__)
#if __has_builtin(__builtin_amdgcn_tanhf)
    return __builtin_amdgcn_tanhf(x);
#else
    float r;
    // TRANS hazard: 1 independent op required before the result is consumed;
    // the compiler can't see inside the asm, so pad with v_nop.
    asm("v_tanh_f32 %0, %1\n\tv_nop" : "=v"(r) : "v"(x));
    return r;
#endif
#else
    return tanhf(x);
#endif
}

__device__ __forceinline__ float dev_exp2(float x) {
#if defined(__AMDGCN__)
    return __builtin_amdgcn_exp2f(x);   // v_exp_f32
#else
    return exp2f(x);
#endif
}

// ---------------------------------------------------------------------------
// Elementwise Lee-oscillator recurrence, 4 elements per thread (b128 ld/st).
// Per element:   decay = exp(-K x^2), w = tanh(x)  (hoisted)
//   49x:  s  = fma(-0.5, z, 0.5*x)
//         u' = tanh(fma(0.6, u - v, s))
//         v' = tanh(fma(0.6, u + v, s))
//         z' = fma(v' - u', decay, w)
// 4 independent lanes/thread x 7-deep unroll: plenty of independent VALU to
// co-issue against the TRANS-class tanh ops and amortize loop SALU.
// ---------------------------------------------------------------------------
__global__ __launch_bounds__(256) void lee_tanh_kernel(
    const float* __restrict__ x, float* __restrict__ out, int n4, int n) {

    const int tid    = blockIdx.x * blockDim.x + threadIdx.x;
    const int stride = gridDim.x * blockDim.x;

    const float4* __restrict__ x4 = reinterpret_cast<const float4*>(x);
    float4* __restrict__       o4 = reinterpret_cast<float4*>(out);

    for (int i = tid; i < n4; i += stride) {
        float4 xv = x4[i];                              // global_load_b128
        float xs[4] = {xv.x, xv.y, xv.z, xv.w};

        float u[4], v[4], z[4], hx[4], decay[4], w[4];
#pragma unroll
        for (int j = 0; j < 4; ++j) {
            const float xx = xs[j];
            decay[j] = dev_exp2((xx * xx) * (-LEE_K * LOG2E_F)); // exp(-K x^2)
            w[j]     = dev_tanh(xx);
            hx[j]    = 0.5f * xx;
            u[j] = 0.2f; v[j] = 0.0f; z[j] = 0.2f;
        }

#pragma unroll 7
        for (int it = 0; it < LEE_ITERS; ++it) {
#pragma unroll
            for (int j = 0; j < 4; ++j) {
                const float s  = __builtin_fmaf(-0.5f, z[j], hx[j]);
                const float tu = __builtin_fmaf(0.6f, u[j] - v[j], s);
                const float tv = __builtin_fmaf(0.6f, u[j] + v[j], s);
                u[j] = dev_tanh(tu);
                v[j] = dev_tanh(tv);
                z[j] = __builtin_fmaf(v[j] - u[j], decay[j], w[j]);
            }
        }

        float4 zv = make_float4(z[0], z[1], z[2], z[3]);
        o4[i] = zv;                                      // global_store_b128
    }

    // Scalar tail (n not divisible by 4) — absent for the reference shape.
    for (int i = n4 * 4 + tid; i < n; i += stride) {
        const float xx    = x[i];
        const float decay = dev_exp2((xx * xx) * (-LEE_K * LOG2E_F));
        const float w     = dev_tanh(xx);
        const float hx    = 0.5f * xx;
        float u = 0.2f, v = 0.0f, z = 0.2f;
        for (int it = 0; it < LEE_ITERS; ++it) {
            const float s  = __builtin_fmaf(-0.5f, z, hx);
            const float tu = __builtin_fmaf(0.6f, u - v, s);
            const float tv = __builtin_fmaf(0.6f, u + v, s);
            u = dev_tanh(tu);
            v = dev_tanh(tv);
            z = __builtin_fmaf(v - u, decay, w);
        }
        out[i] = z;
    }
}

extern "C" void kernel_launch(void* const* d_in, const int* in_sizes, int n_in,
                              void* d_out, int out_size, void* d_ws, size_t ws_size,
                              hipStream_t stream) {
    (void)n_in; (void)d_ws; (void)ws_size;
    const float* x = (const float*)d_in[0];
    float* out     = (float*)d_out;

    const int n  = in_sizes[0];       // 4*1024*512 = 2,097,152
    const int n4 = n / 4;             // 524,288 float4 work-items

    const int block = 256;                       // 8 wave32 waves per block
    int grid = (n4 + block - 1) / block;         // exact cover: 2048 blocks
    if (grid < 1) grid = 1;

    lee_tanh_kernel<<<grid, block, 0, stream>>>(x, out, n4, n);
}